// TwoGNNDirectScoutPolicy_54631984005471
// MI455X (gfx1250) — compile-verified
//
#include <hip/hip_runtime.h>
#include <hip/hip_bf16.h>

// ---------------------------------------------------------------------------
// TwoGNNDirectScoutPolicy for MI455X (gfx1250, wave32)
//
// Pipeline per branch (move, look):
//   for l in 0..1:
//     xlr = h @ [Wl_l | Wr_l]        (WMMA f32 16x16x4, K=8 via 2 chained ops)
//     3 edge passes: segment-max (flipped-uint atomicMax), exp/den (atomicAdd),
//                    alpha-weighted aggregation into h_next (atomicAdd)
//     h_next += bias_l
//   mean-pool over nodes + 2-layer tanh MLP -> agg[B,32]   (1 block / batch)
//   per-agent-edge 3-layer scorer + ordered scatter into logits
//
// Everything stays in d_ws (~41 MB, L2-resident on MI455X's 192 MB L2).
// ---------------------------------------------------------------------------

namespace {

constexpr int B   = 128;
constexpr int N   = 2000;
constexpr int DEG = 16;
constexpr int F   = 8;
constexpr int NUM_ACT = 11;
constexpr int L   = 2;
constexpr int H   = 4;
constexpr int CH  = 2;          // channels per head; H*CH == F
constexpr int E0  = N * DEG;    // 32000 real edges
constexpr int E   = E0 + N;     // + self loops = 34000
constexpr int R   = B * N;      // 256000 node rows
constexpr int TILES = R / 16;   // 16000 WMMA row-tiles (exact)

typedef __attribute__((ext_vector_type(2))) float v2f;
typedef __attribute__((ext_vector_type(8))) float v8f;

// order-preserving float <-> uint mapping for atomicMax-based segment max
static __device__ __forceinline__ unsigned fflip(float f) {
  unsigned u = __float_as_uint(f);
  return (u & 0x80000000u) ? ~u : (u | 0x80000000u);
}
static __device__ __forceinline__ float funflip(unsigned u) {
  return __uint_as_float((u & 0x80000000u) ? (u ^ 0x80000000u) : ~u);
}

// ---------------------------------------------------------------------------
// Node transform: xlr[r, 0:8] = h[r] @ Wl,  xlr[r, 8:16] = h[r] @ Wr
// GEMM M=R, K=8, N=16 using V_WMMA_F32_16X16X4_F32 (2 chained ops for K=8).
// One wave handles one 16-row tile. EXEC is all-ones (wave-uniform guard).
// ---------------------------------------------------------------------------
__global__ void k_linear(const float* __restrict__ hin,   // [R, 8]
                         const float* __restrict__ Wl,    // [8, 8]
                         const float* __restrict__ Wr,    // [8, 8]
                         float* __restrict__ xlr)         // [R, 16]
{
  const int wave = (blockIdx.x * blockDim.x + threadIdx.x) >> 5;
  const int lane = threadIdx.x & 31;
  if (wave >= TILES) return;                 // wave-uniform
  const int half = lane >> 4;                // 0: K={0,1}, 1: K={2,3}
  const int l16  = lane & 15;
  const int kb   = half * 2;

  // A tile (16x4 f32): lanes 0-15 rows 0-15 K={0,1}; lanes 16-31 K={2,3}
  const float* hr = hin + (size_t)(wave * 16 + l16) * F;
  v2f a0, a1;
  a0.x = hr[kb + 0]; a0.y = hr[kb + 1];      // K 0..3
  a1.x = hr[kb + 4]; a1.y = hr[kb + 5];      // K 4..7

  // B tile (4x16): column l16; cols 0-7 from Wl, 8-15 from Wr
  const float* Wsel = (l16 < 8) ? Wl : Wr;
  const int n8 = l16 & 7;
  v2f b0, b1;
  b0.x = Wsel[(kb + 0) * 8 + n8]; b0.y = Wsel[(kb + 1) * 8 + n8];
  b1.x = Wsel[(kb + 4) * 8 + n8]; b1.y = Wsel[(kb + 5) * 8 + n8];

  v8f c = {};
  c = __builtin_amdgcn_wmma_f32_16x16x4_f32(false, a0, false, b0, (short)0, c, false, false);
  c = __builtin_amdgcn_wmma_f32_16x16x4_f32(false, a1, false, b1, (short)0, c, false, false);

  // D layout: VGPR v, lanes 0-15 -> (M=v, N=lane); lanes 16-31 -> (M=8+v, N=lane-16)
  const int mbase = wave * 16 + half * 8;
#pragma unroll
  for (int v = 0; v < 8; ++v)
    xlr[(size_t)(mbase + v) * 16 + l16] = c[v];
}

// ---------------------------------------------------------------------------
// Edge passes. grid = (ceil(E/256), B). Edge e >= E0 is the self-loop e-E0.
// ---------------------------------------------------------------------------
static __device__ __forceinline__ void edge_logits(
    const float* __restrict__ xlr, int b, int e,
    const int* __restrict__ esrc, const int* __restrict__ edst,
    const float* __restrict__ att,
    int& src, int& dst, float s[H], const float*& xl_src)
{
  if (e < E0) { src = esrc[e]; dst = edst[e]; }
  else        { src = dst = e - E0; }
  const float* xb = xlr + (size_t)b * N * 16;
  const float* xl = xb + (size_t)src * 16;       // first 8 = xl
  const float* xr = xb + (size_t)dst * 16 + 8;   // last 8 = xr
  xl_src = xl;
#pragma unroll
  for (int h = 0; h < H; ++h) {
    float acc = 0.f;
#pragma unroll
    for (int c = 0; c < CH; ++c) {
      float v = xl[h * CH + c] + xr[h * CH + c];
      v = (v > 0.f) ? v : 0.2f * v;              // leaky_relu(0.2)
      acc += v * att[h * CH + c];
    }
    s[h] = acc;
  }
}

__global__ void k_edge_max(const float* __restrict__ xlr,
                           const int* __restrict__ esrc, const int* __restrict__ edst,
                           const float* __restrict__ att,
                           unsigned* __restrict__ emax)   // [B,N,H] flipped
{
  const int e = blockIdx.x * blockDim.x + threadIdx.x;
  if (e >= E) return;
  const int b = blockIdx.y;
  int src, dst; float s[H]; const float* xl;
  edge_logits(xlr, b, e, esrc, edst, att, src, dst, s, xl);
  const size_t base = ((size_t)b * N + dst) * H;
#pragma unroll
  for (int h = 0; h < H; ++h)
    atomicMax(emax + base + h, fflip(s[h]));
}

__global__ void k_edge_den(const float* __restrict__ xlr,
                           const int* __restrict__ esrc, const int* __restrict__ edst,
                           const float* __restrict__ att,
                           const unsigned* __restrict__ emax,
                           float* __restrict__ den)       // [B,N,H]
{
  const int e = blockIdx.x * blockDim.x + threadIdx.x;
  if (e >= E) return;
  const int b = blockIdx.y;
  int src, dst; float s[H]; const float* xl;
  edge_logits(xlr, b, e, esrc, edst, att, src, dst, s, xl);
  const size_t base = ((size_t)b * N + dst) * H;
#pragma unroll
  for (int h = 0; h < H; ++h)
    atomicAdd(den + base + h, expf(s[h] - funflip(emax[base + h])));
}

__global__ void k_edge_agg(const float* __restrict__ xlr,
                           const int* __restrict__ esrc, const int* __restrict__ edst,
                           const float* __restrict__ att,
                           const unsigned* __restrict__ emax,
                           const float* __restrict__ den,
                           float* __restrict__ hnext)     // [B,N,8] (pre-zeroed)
{
  const int e = blockIdx.x * blockDim.x + threadIdx.x;
  if (e >= E) return;
  const int b = blockIdx.y;
  int src, dst; float s[H]; const float* xl;
  edge_logits(xlr, b, e, esrc, edst, att, src, dst, s, xl);
  const size_t base = ((size_t)b * N + dst) * H;
  float* outp = hnext + ((size_t)b * N + dst) * F;
#pragma unroll
  for (int h = 0; h < H; ++h) {
    const float ez = expf(s[h] - funflip(emax[base + h]));
    const float w  = ez / (den[base + h] + 1e-16f);
    atomicAdd(outp + h * CH + 0, w * xl[h * CH + 0]);
    atomicAdd(outp + h * CH + 1, w * xl[h * CH + 1]);
  }
}

__global__ void k_bias(float* __restrict__ hnext, const float* __restrict__ bias) {
  const int t = blockIdx.x * blockDim.x + threadIdx.x;   // over R*8
  if (t >= R * F) return;
  hnext[t] += bias[t & (F - 1)];
}

// ---------------------------------------------------------------------------
// Mean over nodes + global MLP (tanh x2): one block (256 thr) per batch elem.
// ---------------------------------------------------------------------------
__global__ void k_meanagg(const float* __restrict__ feat,   // [B,N,8]
                          const float* __restrict__ gW1, const float* __restrict__ gb1,
                          const float* __restrict__ gW2, const float* __restrict__ gb2,
                          float* __restrict__ agg)          // [B,32]
{
  __shared__ float sdat[256];
  __shared__ float mean[8];
  __shared__ float g1[16];
  const int b = blockIdx.x, t = threadIdx.x;
  const int j = t & 7, n0 = t >> 3;
  float acc = 0.f;
  for (int n = n0; n < N; n += 32)
    acc += feat[((size_t)b * N + n) * F + j];
  sdat[t] = acc;
  __syncthreads();
  for (int off = 128; off >= 8; off >>= 1) {
    if (t < off) sdat[t] += sdat[t + off];
    __syncthreads();
  }
  if (t < 8) mean[t] = sdat[t] * (1.0f / (float)N);
  __syncthreads();
  if (t < 16) {
    float a = gb1[t];
    for (int k = 0; k < 8; ++k) a += mean[k] * gW1[k * 16 + t];
    g1[t] = tanhf(a);
  }
  __syncthreads();
  if (t < 32) {
    float a = gb2[t];
    for (int k = 0; k < 16; ++k) a += g1[k] * gW2[k * 32 + t];
    agg[b * 32 + t] = tanhf(a);
  }
}

// ---------------------------------------------------------------------------
// Per-agent-edge scorer + ordered scatter (last-write-wins, like .at[].set).
// One wave per batch element; lanes 0..15 score the DEG agent edges.
// ---------------------------------------------------------------------------
__global__ void k_scores(const float* __restrict__ feat,   // [B,N,8]
                         const float* __restrict__ agg,    // [B,32]
                         const int* __restrict__ esrc, const int* __restrict__ edst,
                         const int* __restrict__ agent, const int* __restrict__ actions,
                         const float* __restrict__ eW1, const float* __restrict__ eb1,
                         const float* __restrict__ eW2, const float* __restrict__ eb2,
                         const float* __restrict__ eW3, const float* __restrict__ eb3,
                         float* __restrict__ out, int colBase, int isLook)
{
  __shared__ float sc[DEG];
  const int b = blockIdx.x, t = threadIdx.x;
  if (t < DEG) {
    const int eidx = agent[b] * DEG + t;
    const int s = esrc[eidx], d = edst[eidx];
    const float* sf = feat + ((size_t)b * N + s) * F;
    const float* tf = feat + ((size_t)b * N + d) * F;
    const float* g  = agg + b * 32;
    float in48[48];
#pragma unroll
    for (int k = 0; k < 8; ++k)  in48[k]      = sf[k];
#pragma unroll
    for (int k = 0; k < 8; ++k)  in48[8 + k]  = tf[k];
#pragma unroll
    for (int k = 0; k < 32; ++k) in48[16 + k] = g[k];
    float h1[8];
#pragma unroll
    for (int i = 0; i < 8; ++i) {
      float a = eb1[i];
      for (int k = 0; k < 48; ++k) a += in48[k] * eW1[k * 8 + i];
      h1[i] = tanhf(a);
    }
    float h2[4];
#pragma unroll
    for (int i = 0; i < 4; ++i) {
      float a = eb2[i];
      for (int k = 0; k < 8; ++k) a += h1[k] * eW2[k * 4 + i];
      h2[i] = tanhf(a);
    }
    float v = eb3[0];
#pragma unroll
    for (int k = 0; k < 4; ++k) v += h2[k] * eW3[k];
    sc[t] = v;
  }
  __syncthreads();
  if (t == 0) {
    float lg[NUM_ACT];
#pragma unroll
    for (int k = 0; k < NUM_ACT; ++k) lg[k] = 0.f;
    lg[NUM_ACT - 1] = 1.f;
    for (int j = 0; j < DEG; ++j) {          // ascending order: last-write-wins
      const int a = actions[b * DEG + j];
      if (isLook) { if (a != 0) lg[a] = sc[j]; }   // a==0 routed to dummy col
      else        { lg[a] = sc[j]; }
    }
#pragma unroll
    for (int k = 0; k < NUM_ACT; ++k)
      out[b * (2 * NUM_ACT) + colBase + k] = lg[k];
  }
}

} // namespace

// ---------------------------------------------------------------------------
// Host launcher. Input order = setup_inputs() insertion order, nested dicts
// flattened in insertion order:
//   0: x | 1-4: move_gat {Wl,Wr,att,bias} | 5-8: look_gat
//   9-18: move_mlp {gW1,gb1,gW2,gb2,eW1,eb1,eW2,eb2,eW3,eb3} | 19-28: look_mlp
//   29: edge_index [2,E0] | 30: agent_nodes [B] | 31: actions [B,DEG]
// ---------------------------------------------------------------------------
extern "C" void kernel_launch(void* const* d_in, const int* in_sizes, int n_in,
                              void* d_out, int out_size, void* d_ws, size_t ws_size,
                              hipStream_t stream) {
  (void)in_sizes; (void)n_in; (void)out_size; (void)ws_size;

  const float* x = (const float*)d_in[0];
  struct Gat { const float *Wl, *Wr, *att, *bias; };
  struct Mlp { const float *gW1, *gb1, *gW2, *gb2, *eW1, *eb1, *eW2, *eb2, *eW3, *eb3; };
  Gat gat[2];
  Mlp mlp[2];
  for (int br = 0; br < 2; ++br) {
    const int gb = 1 + br * 4;
    gat[br] = { (const float*)d_in[gb + 0], (const float*)d_in[gb + 1],
                (const float*)d_in[gb + 2], (const float*)d_in[gb + 3] };
    const int mb = 9 + br * 10;
    mlp[br] = { (const float*)d_in[mb + 0], (const float*)d_in[mb + 1],
                (const float*)d_in[mb + 2], (const float*)d_in[mb + 3],
                (const float*)d_in[mb + 4], (const float*)d_in[mb + 5],
                (const float*)d_in[mb + 6], (const float*)d_in[mb + 7],
                (const float*)d_in[mb + 8], (const float*)d_in[mb + 9] };
  }
  const int* edge_index = (const int*)d_in[29];
  const int* esrc  = edge_index;        // [E0]
  const int* edst  = edge_index + E0;   // [E0]
  const int* agent = (const int*)d_in[30];
  const int* acts  = (const int*)d_in[31];
  float* out = (float*)d_out;

  // workspace carve-up (~41 MB)
  char* ws = (char*)d_ws;
  float*    hA   = (float*)ws;    ws += sizeof(float) * (size_t)R * F;    // 8.2 MB
  float*    hB   = (float*)ws;    ws += sizeof(float) * (size_t)R * F;    // 8.2 MB
  float*    xlr  = (float*)ws;    ws += sizeof(float) * (size_t)R * 16;   // 16.4 MB
  unsigned* emax = (unsigned*)ws; ws += sizeof(unsigned) * (size_t)R * H; // 4.1 MB
  float*    den  = (float*)ws;    ws += sizeof(float) * (size_t)R * H;    // 4.1 MB
  float*    agg  = (float*)ws;    ws += sizeof(float) * (size_t)B * 32;

  const dim3 egrid((E + 255) / 256, B);

  for (int br = 0; br < 2; ++br) {
    const float* hin = x;
    for (int l = 0; l < L; ++l) {
      float* hnext = (l == 0) ? hA : hB;
      hipMemsetAsync(hnext, 0, sizeof(float) * (size_t)R * F, stream);
      hipMemsetAsync(emax, 0, sizeof(unsigned) * (size_t)R * H, stream);  // flipped -inf
      hipMemsetAsync(den,  0, sizeof(float) * (size_t)R * H, stream);

      k_linear<<<TILES / 8, 256, 0, stream>>>(hin, gat[br].Wl + l * 64,
                                              gat[br].Wr + l * 64, xlr);
      const float* attl = gat[br].att + l * 8;
      k_edge_max<<<egrid, 256, 0, stream>>>(xlr, esrc, edst, attl, emax);
      k_edge_den<<<egrid, 256, 0, stream>>>(xlr, esrc, edst, attl, emax, den);
      k_edge_agg<<<egrid, 256, 0, stream>>>(xlr, esrc, edst, attl, emax, den, hnext);
      k_bias<<<(R * F + 255) / 256, 256, 0, stream>>>(hnext, gat[br].bias + l * 8);
      hin = hnext;
    }
    const float* feat = hB;
    k_meanagg<<<B, 256, 0, stream>>>(feat, mlp[br].gW1, mlp[br].gb1,
                                     mlp[br].gW2, mlp[br].gb2, agg);
    k_scores<<<B, 32, 0, stream>>>(feat, agg, esrc, edst, agent, acts,
                                   mlp[br].eW1, mlp[br].eb1, mlp[br].eW2, mlp[br].eb2,
                                   mlp[br].eW3, mlp[br].eb3,
                                   out, br * NUM_ACT, br);
  }
}